// SP_GNNStage_53609781789202
// MI455X (gfx1250) — compile-verified
//
#include <hip/hip_runtime.h>
#include <hip/hip_bf16.h>

typedef float v2f __attribute__((ext_vector_type(2)));
typedef float v8f __attribute__((ext_vector_type(8)));

#define NN      50000
#define NE      800000
#define DD      64
#define KREL    3
#define NLAYERS 3

// ---------------- degree pipeline ----------------

__global__ void deg_init_kernel(float* __restrict__ deg) {
    int i = blockIdx.x * blockDim.x + threadIdx.x;
    if (i < KREL * NN) deg[i] = 1.0f;   // self-loop included
}

__global__ void deg_count_kernel(const int* __restrict__ ei,
                                 const int* __restrict__ attr,
                                 float* __restrict__ deg) {
    int e = blockIdx.x * blockDim.x + threadIdx.x;
    if (e >= NE) return;
    int d = ei[NE + e];          // dst row of edge_index
    int k = attr[e];
    unsafeAtomicAdd(&deg[k * NN + d], 1.0f);   // exact: counts < 2^24
}

__global__ void deg_finish_kernel(const float* __restrict__ deg,
                                  float* __restrict__ dinv,
                                  float* __restrict__ deginv) {
    int i = blockIdx.x * blockDim.x + threadIdx.x;
    if (i >= KREL * NN) return;
    float dv  = deg[i];
    dinv[i]   = 1.0f / sqrtf(dv);
    deginv[i] = 1.0f / dv;
}

// ---------------- weight swizzle ----------------
// Pair-interleave W so each WMMA B-operand pair {W[r][c], W[r+1][c]} is an
// 8-byte-contiguous, 8-byte-aligned LDS load: one ds_load_b64 per fragment.
// Wsz[tk][((row>>1)*DD + col)*2 + (row&1)] = W[tk][row][col]

__global__ void wswz_kernel(const float* __restrict__ W,
                            float* __restrict__ Wsz) {
    int i = blockIdx.x * blockDim.x + threadIdx.x;
    if (i >= NLAYERS * KREL * DD * DD) return;
    int col = i & 63;
    int row = (i >> 6) & 63;
    int tk  = i >> 12;                       // t*KREL + k
    Wsz[(size_t)tk * DD * DD + (((row >> 1) * DD + col) << 1) + (row & 1)] = W[i];
}

// ---------------- fp32 WMMA GEMM: xW[k] = x @ W[t,k] ----------------
// Swizzled W[t] (48 KB, all 3 relations) staged once per block into LDS via
// the CDNA5 async-to-LDS path (ASYNCcnt), then each wave computes a
// 16(node-rows) x 64(out-cols) strip for one relation with
// V_WMMA_F32_16X16X4_F32; B fragments are direct ds_load_b64 from LDS.

__global__ __launch_bounds__(256)
void gemm_xw_kernel(const float* __restrict__ x,     // [NN][DD]
                    const float* __restrict__ Wt,    // [KREL][DD][DD] swizzled (layer t)
                    float* __restrict__ xW) {        // [KREL][NN][DD]
    __shared__ __align__(16) float sW[KREL * DD * DD];   // 48 KB

    // ---- async copy swizzled W[t] -> LDS (global_load_async_to_lds_b128) ----
    {
        uint32_t lds_base = (uint32_t)(uintptr_t)&sW[0];   // LDS byte offset (low 32 bits)
        const int nchunks = (KREL * DD * DD) / 4;          // 16B chunks = 3072
        for (int c = threadIdx.x; c < nchunks; c += blockDim.x) {
            uint32_t loff = lds_base + (uint32_t)c * 16u;
            const float* ga = Wt + (size_t)c * 4;
            asm volatile("global_load_async_to_lds_b128 %0, %1, off"
                         :: "v"(loff), "v"(ga)
                         : "memory");
        }
        asm volatile("s_wait_asynccnt 0x0" ::: "memory");
    }
    __syncthreads();

    const int lane    = threadIdx.x & 31;
    const int gwave   = (blockIdx.x * blockDim.x + threadIdx.x) >> 5;
    const int k       = gwave % KREL;
    const int rowtile = gwave / KREL;
    if (rowtile >= NN / 16) return;          // wave-uniform: EXEC stays all-1s
    const int row0 = rowtile * 16;
    const int hl   = lane & 15;              // lane within half
    const int hi   = lane >> 4;              // 0 = lanes 0-15, 1 = lanes 16-31

    const float* Wk = &sW[k * DD * DD];

    v8f acc0 = {}, acc1 = {}, acc2 = {}, acc3 = {};
    for (int kk = 0; kk < DD; kk += 4) {
        // A 16x4 (f32): lanes 0-15 hold M=lane, K=kk..kk+1; lanes 16-31 K=kk+2..kk+3
        // (even element offset -> 8B aligned -> global_load_b64)
        const float* xr = x + (size_t)(row0 + hl) * DD + kk + hi * 2;
        v2f a = *(const v2f*)xr;
        // B 4x16 (f32): v0 = rows {kk | kk+2}, v1 = rows {kk+1 | kk+3}, N = col.
        // Swizzled layout: rows (kk+hi*2, kk+hi*2+1) at col are contiguous.
        const int pbase = ((kk >> 1) + hi) * DD;
        v2f b0 = *(const v2f*)&Wk[(pbase + hl)      * 2];
        v2f b1 = *(const v2f*)&Wk[(pbase + hl + 16) * 2];
        v2f b2 = *(const v2f*)&Wk[(pbase + hl + 32) * 2];
        v2f b3 = *(const v2f*)&Wk[(pbase + hl + 48) * 2];
        acc0 = __builtin_amdgcn_wmma_f32_16x16x4_f32(false, a, false, b0, (short)0, acc0, false, false);
        acc1 = __builtin_amdgcn_wmma_f32_16x16x4_f32(false, a, false, b1, (short)0, acc1, false, false);
        acc2 = __builtin_amdgcn_wmma_f32_16x16x4_f32(false, a, false, b2, (short)0, acc2, false, false);
        acc3 = __builtin_amdgcn_wmma_f32_16x16x4_f32(false, a, false, b3, (short)0, acc3, false, false);
    }

    // C/D layout: VGPR j <-> row (j + hi*8), col = hl (per 16-wide tile)
    float* out = xW + ((size_t)k * NN + row0) * DD;
#pragma unroll
    for (int j = 0; j < 8; ++j) {
        int m = j + hi * 8;
        out[(size_t)m * DD +  0 + hl] = acc0[j];
        out[(size_t)m * DD + 16 + hl] = acc1[j];
        out[(size_t)m * DD + 32 + hl] = acc2[j];
        out[(size_t)m * DD + 48 + hl] = acc3[j];
    }
}

// ---------------- per-layer aggregation ----------------

__global__ void agg_zero_kernel(float* __restrict__ agg) {
    int i = blockIdx.x * blockDim.x + threadIdx.x;
    if (i < NN * DD) agg[i] = 0.0f;
}

// 64 threads per edge (one per feature); fp32 hardware atomics into L2-resident agg
__global__ __launch_bounds__(256)
void edge_scatter_kernel(const int* __restrict__ ei,
                         const int* __restrict__ attr,
                         const float* __restrict__ dinv,
                         const float* __restrict__ xW,
                         float* __restrict__ agg) {
    int t = blockIdx.x * blockDim.x + threadIdx.x;
    int e = t >> 6;
    if (e >= NE) return;
    int j = t & 63;
    int s = ei[e];
    int d = ei[NE + e];
    int k = attr[e];
    float w = dinv[k * NN + s] * dinv[k * NN + d];
    float v = xW[((size_t)k * NN + s) * DD + j] * w;
    unsafeAtomicAdd(&agg[(size_t)d * DD + j], v);
}

// self-loop + bias + residual + ReLU + L2 normalize; 32 lanes per node (2 feats/lane)
__global__ __launch_bounds__(256)
void node_update_kernel(const float* __restrict__ xin,
                        const float* __restrict__ xW,
                        const float* __restrict__ agg,
                        const float* __restrict__ deginv,
                        const float* __restrict__ bt,    // [KREL][DD] (layer t)
                        float* __restrict__ xout) {
    int lane = threadIdx.x & 31;
    int node = (blockIdx.x * blockDim.x + threadIdx.x) >> 5;
    if (node >= NN) return;
    int j0 = lane, j1 = lane + 32;

    float b0 = bt[j0] + bt[DD + j0] + bt[2 * DD + j0];
    float b1 = bt[j1] + bt[DD + j1] + bt[2 * DD + j1];
    float a0 = agg[(size_t)node * DD + j0] + b0;
    float a1 = agg[(size_t)node * DD + j1] + b1;
#pragma unroll
    for (int k = 0; k < KREL; ++k) {
        float di = deginv[k * NN + node];
        const float* xr = xW + ((size_t)k * NN + node) * DD;
        a0 += di * xr[j0];
        a1 += di * xr[j1];
    }
    float x0 = xin[(size_t)node * DD + j0] + fmaxf(a0, 0.0f);
    float x1 = xin[(size_t)node * DD + j1] + fmaxf(a1, 0.0f);

    float ss = x0 * x0 + x1 * x1;
#pragma unroll
    for (int off = 16; off > 0; off >>= 1) ss += __shfl_xor(ss, off, 32);
    float inv = 1.0f / fmaxf(sqrtf(ss), 1e-12f);

    xout[(size_t)node * DD + j0] = x0 * inv;
    xout[(size_t)node * DD + j1] = x1 * inv;
}

// ---------------- host ----------------

extern "C" void kernel_launch(void* const* d_in, const int* in_sizes, int n_in,
                              void* d_out, int out_size, void* d_ws, size_t ws_size,
                              hipStream_t stream) {
    const float* x  = (const float*)d_in[0];   // [NN, DD]
    const int*   ei = (const int*)d_in[1];     // [2, NE]
    const int*   ea = (const int*)d_in[2];     // [NE]
    const float* W  = (const float*)d_in[3];   // [NLAYERS, KREL, DD, DD]
    const float* b  = (const float*)d_in[4];   // [NLAYERS, KREL, DD]
    float* out = (float*)d_out;

    char* ws = (char*)d_ws;
    float* deg    = (float*)ws;  ws += sizeof(float) * (size_t)KREL * NN;
    float* dinv   = (float*)ws;  ws += sizeof(float) * (size_t)KREL * NN;
    float* deginv = (float*)ws;  ws += sizeof(float) * (size_t)KREL * NN;
    float* wsz    = (float*)ws;  ws += sizeof(float) * (size_t)NLAYERS * KREL * DD * DD;
    float* xW     = (float*)ws;  ws += sizeof(float) * (size_t)KREL * NN * DD;
    float* agg    = (float*)ws;  ws += sizeof(float) * (size_t)NN * DD;
    float* xcur   = (float*)ws;  ws += sizeof(float) * (size_t)NN * DD;

    // degrees / normalization constants (recomputed every call: deterministic)
    deg_init_kernel<<<(KREL * NN + 255) / 256, 256, 0, stream>>>(deg);
    deg_count_kernel<<<(NE + 255) / 256, 256, 0, stream>>>(ei, ea, deg);
    deg_finish_kernel<<<(KREL * NN + 255) / 256, 256, 0, stream>>>(deg, dinv, deginv);

    // weight swizzle (all layers, once per call)
    wswz_kernel<<<(NLAYERS * KREL * DD * DD + 255) / 256, 256, 0, stream>>>(W, wsz);

    hipMemcpyAsync(xcur, x, sizeof(float) * (size_t)NN * DD,
                   hipMemcpyDeviceToDevice, stream);

    const int gemm_waves  = (NN / 16) * KREL;           // 9375 waves
    const int gemm_blocks = (gemm_waves + 7) / 8;       // 8 waves / 256-thread block

    for (int t = 0; t < NLAYERS; ++t) {
        const float* Wt = wsz + (size_t)t * KREL * DD * DD;
        const float* bt = b + (size_t)t * KREL * DD;
        float* xo = (t == NLAYERS - 1) ? out : xcur;

        gemm_xw_kernel<<<gemm_blocks, 256, 0, stream>>>(xcur, Wt, xW);
        agg_zero_kernel<<<(NN * DD + 255) / 256, 256, 0, stream>>>(agg);
        edge_scatter_kernel<<<(NE * 64 + 255) / 256, 256, 0, stream>>>(ei, ea, dinv, xW, agg);
        node_update_kernel<<<(NN * 32 + 255) / 256, 256, 0, stream>>>(xcur, xW, agg, deginv, bt, xo);
    }
}